// AGNN_68951404970012
// MI455X (gfx1250) — compile-verified
//
#include <hip/hip_runtime.h>
#include <math.h>

typedef float v2f __attribute__((ext_vector_type(2)));
typedef float v8f __attribute__((ext_vector_type(8)));

#define D_FEAT 64
#define NORM_EPS 1e-12f

// ---------------------------------------------------------------- zero
__global__ void agnn_zero(float* __restrict__ p, int n) {
    int i = blockIdx.x * blockDim.x + threadIdx.x;
    if (i < n) p[i] = 0.0f;
}

// ---------------------------------------------------------------- normalize
// One wave32 per node; each lane owns 2 consecutive features (32*2 = 64).
__global__ void agnn_normalize(const float* __restrict__ h,
                               float* __restrict__ nh, int n_nodes) {
    int wave = (blockIdx.x * blockDim.x + threadIdx.x) >> 5;
    int lane = threadIdx.x & 31;
    if (wave >= n_nodes) return;
    const float* row = h + (size_t)wave * D_FEAT + 2 * lane;
    float x0 = row[0], x1 = row[1];
    float ss = x0 * x0 + x1 * x1;
#pragma unroll
    for (int off = 16; off >= 1; off >>= 1)
        ss += __shfl_xor(ss, off, 32);
    float inv = 1.0f / fmaxf(sqrtf(ss), NORM_EPS);
    float* orow = nh + (size_t)wave * D_FEAT + 2 * lane;
    orow[0] = x0 * inv;
    orow[1] = x1 * inv;
}

// ---------------------------------------------------------------- edge logits (WMMA)
// One wave32 per 16 edges. A = nh[src rows] (16x64), B^T = nh[dst rows];
// diag(A x B) via 16 chained V_WMMA_F32_16X16X4_F32 steps gives the 16 cosines.
// f32 A 16x4 layout: lane L holds row M=L&15, cols 4k + 2*(L>>4) + {0,1} -> the
// per-lane load pattern is identical for A and B (one float2 per step each).
__global__ void agnn_edge_logits(const float* __restrict__ nh,
                                 const int* __restrict__ src,
                                 const int* __restrict__ dst,
                                 const float* __restrict__ beta, int layer,
                                 float* __restrict__ w,
                                 float* __restrict__ den,
                                 int n_edges) {
    int wave = blockIdx.x * (blockDim.x >> 5) + (threadIdx.x >> 5);
    int lane = threadIdx.x & 31;
    int n_full = n_edges >> 4;

    if (wave < n_full) {
        // --- WMMA path: whole-wave uniform branch, EXEC all ones ---
        int base = wave << 4;
        int r    = lane & 15;
        int hi   = lane >> 4;
        int e    = base + r;
        size_t srow = (size_t)src[e];
        size_t drow = (size_t)dst[e];
        const float* ap = nh + srow * D_FEAT + 2 * hi;
        const float* bp = nh + drow * D_FEAT + 2 * hi;

        v2f a[16], b[16];
#pragma unroll
        for (int k = 0; k < 16; ++k) {
            a[k] = *(const v2f*)(ap + 4 * k);
            b[k] = *(const v2f*)(bp + 4 * k);
        }

        v8f c = {0.f, 0.f, 0.f, 0.f, 0.f, 0.f, 0.f, 0.f};
#pragma unroll
        for (int k = 0; k < 16; ++k) {
            c = __builtin_amdgcn_wmma_f32_16x16x4_f32(
                    /*neg_a=*/false, a[k], /*neg_b=*/false, b[k],
                    /*c_mod=*/(short)0, c, /*reuse_a=*/false, /*reuse_b=*/false);
        }

        // Diagonal extraction: D[i][i] lives at (vgpr=i, lane=i) for i<8 and
        // (vgpr=i-8, lane=i+16) for i>=8. Lanes with (lane%16)<8 each take
        // one edge row = (lane&7) + 8*hi, value = c[lane&7], which is D[row][row] only in lanes 0-7.
        int s = lane & 7;
        float dg = c[0];
        dg = (s == 1) ? c[1] : dg;
        dg = (s == 2) ? c[2] : dg;
        dg = (s == 3) ? c[3] : dg;
        dg = (s == 4) ? c[4] : dg;
        dg = (s == 5) ? c[5] : dg;
        dg = (s == 6) ? c[6] : dg;
        dg = (s == 7) ? c[7] : dg;

        if ((lane & 15) < 8) {
            int row = (lane & 7) + (hi ? 8 : 0);
            int ee  = base + row;
            // softmax without max-shift: |beta*cos| <= |beta|, exp is safe and
            // exp(e)/sum exp(e) == exp(e-m)/sum exp(e-m) exactly in math.
            float wv = expf(beta[layer] * dg);
            w[ee] = wv;
            atomicAdd(&den[dst[ee]], wv);
        }
    } else if (wave == n_full) {
        // --- scalar tail (no WMMA; per-lane predication is fine here) ---
        int base = n_full << 4;
        int rem  = n_edges - base;
        if (lane < rem) {
            int e = base + lane;
            const float* ar = nh + (size_t)src[e] * D_FEAT;
            const float* br = nh + (size_t)dst[e] * D_FEAT;
            float acc = 0.f;
#pragma unroll
            for (int k = 0; k < D_FEAT; ++k) acc = fmaf(ar[k], br[k], acc);
            float wv = expf(beta[layer] * acc);
            w[e] = wv;
            atomicAdd(&den[dst[e]], wv);
        }
    }
}

// ---------------------------------------------------------------- aggregate
// 64 threads per edge: acc[dst, f] += h[src, f] * (w[e] / den[dst]).
__global__ void agnn_aggregate(const float* __restrict__ h,
                               const int* __restrict__ src,
                               const int* __restrict__ dst,
                               const float* __restrict__ w,
                               const float* __restrict__ den,
                               float* __restrict__ acc,
                               int n_edges) {
    long long t = (long long)blockIdx.x * blockDim.x + threadIdx.x;
    int e = (int)(t >> 6);
    if (e >= n_edges) return;
    int f = (int)(t & 63);
    int d = dst[e];
    float coef = w[e] / den[d];
    float v = h[(size_t)src[e] * D_FEAT + f] * coef;
    atomicAdd(&acc[(size_t)d * D_FEAT + f], v);
}

// ---------------------------------------------------------------- relu
__global__ void agnn_relu(const float* __restrict__ a, float* __restrict__ o, int n) {
    int i = blockIdx.x * blockDim.x + threadIdx.x;
    if (i < n) o[i] = fmaxf(a[i], 0.0f);
}

// ---------------------------------------------------------------- launch
extern "C" void kernel_launch(void* const* d_in, const int* in_sizes, int n_in,
                              void* d_out, int out_size, void* d_ws, size_t ws_size,
                              hipStream_t stream) {
    const float* x    = (const float*)d_in[0];
    const int*   src  = (const int*)d_in[1];
    const int*   dst  = (const int*)d_in[2];
    const float* beta = (const float*)d_in[3];

    const int n_nodes = in_sizes[0] / D_FEAT;
    const int n_edges = in_sizes[1];
    const int nfeat   = n_nodes * D_FEAT;

    // Carve workspace (aligned to 256B).
    char* ws = (char*)d_ws;
    size_t off = 0;
    auto carve = [&](size_t bytes) -> float* {
        float* p = (float*)(ws + off);
        off += (bytes + 255) & ~(size_t)255;
        return p;
    };
    float* nh   = carve((size_t)nfeat * sizeof(float));
    float* hb   = carve((size_t)nfeat * sizeof(float));   // h after layer 1
    float* acc  = carve((size_t)nfeat * sizeof(float));
    float* wbuf = carve((size_t)n_edges * sizeof(float));
    float* den  = carve((size_t)n_nodes * sizeof(float));
    (void)ws_size;

    const dim3 B(256);
    const int gF    = (nfeat + 255) / 256;
    const int gN    = (n_nodes + 255) / 256;
    const int gNorm = (n_nodes * 32 + 255) / 256;                 // 8 nodes/block
    const int nWaves = n_edges / 16 + ((n_edges & 15) ? 1 : 0);   // 16 edges/wave
    const int gEdge = (nWaves + 7) / 8;                           // 8 waves/block
    const long long aggT = (long long)n_edges * 64;
    const int gAgg  = (int)((aggT + 255) / 256);

    auto layer = [&](const float* hin, float* hout, int li) {
        agnn_zero<<<gN, B, 0, stream>>>(den, n_nodes);
        agnn_zero<<<gF, B, 0, stream>>>(acc, nfeat);
        agnn_normalize<<<gNorm, B, 0, stream>>>(hin, nh, n_nodes);
        agnn_edge_logits<<<gEdge, B, 0, stream>>>(nh, src, dst, beta, li,
                                                  wbuf, den, n_edges);
        agnn_aggregate<<<gAgg, B, 0, stream>>>(hin, src, dst, wbuf, den, acc,
                                               n_edges);
        agnn_relu<<<gF, B, 0, stream>>>(acc, hout, nfeat);
    };

    layer(x, hb, 0);
    layer(hb, (float*)d_out, 1);
}